// GNNModel_25744033972357
// MI455X (gfx1250) — compile-verified
//
#include <hip/hip_runtime.h>

// ---------------------------------------------------------------------------
// GCN (2-layer) + mean-pool + FC for MI455X (gfx1250, wave32, WMMA).
//
// Design rationale (roofline, no runtime feedback):
//  * Dense GEMMs (50000x3x128, 50000x128x64) -> V_WMMA_F32_16X16X4_F32
//    (fp32 in/out, matches reference numerics; compute is trivial anyway).
//  * Sparse aggregation: scatter-with-atomics would cost ~115M f32 L2 RMWs,
//    ~100x the streaming cost of the ~100MB of actual traffic. Instead we
//    build a CSR (600K int atomics + deterministic scan) and aggregate by
//    GATHER with direct stores: zero float atomics, bias+ReLU+self-loop
//    fused. Gathered working set (h1 25.6MB, h2 12.8MB, CSR 2.4MB) is
//    L2-resident (192MB), so gathers run at L2 bandwidth, not HBM.
// ---------------------------------------------------------------------------

typedef float v2f __attribute__((ext_vector_type(2)));
typedef float v8f __attribute__((ext_vector_type(8)));

#define N_FEAT1 128
#define N_FEAT2 64

// ---------------- CSR build + normalization ----------------
__global__ void zero_cnt(int* cnt, int N) {
    int i = blockIdx.x * blockDim.x + threadIdx.x;
    if (i < N) cnt[i] = 0;
}

__global__ void count_in(const long long* __restrict__ dst, int* cnt, int E) {
    int e = blockIdx.x * blockDim.x + threadIdx.x;
    if (e < E) atomicAdd(&cnt[(int)dst[e]], 1);
}

// dis[i] = rsqrt(deg_i), deg_i = in_degree + 1 (self loop). deg >= 1 always.
__global__ void make_dis(const int* __restrict__ cnt, float* dis, int N) {
    int i = blockIdx.x * blockDim.x + threadIdx.x;
    if (i < N) dis[i] = rsqrtf(1.0f + (float)cnt[i]);
}

// Deterministic single-block exclusive scan over cnt[0..N) -> rowStart[0..N],
// cursor initialized to rowStart for the CSR fill pass.
__global__ void scan_counts(const int* __restrict__ cnt, int* __restrict__ rowStart,
                            int* __restrict__ cursor, int N) {
    __shared__ int buf[256];
    __shared__ int runningS;
    if (threadIdx.x == 0) runningS = 0;
    __syncthreads();
    for (int base = 0; base < N; base += 256) {
        int idx = base + (int)threadIdx.x;
        int v = (idx < N) ? cnt[idx] : 0;
        buf[threadIdx.x] = v;
        __syncthreads();
        // naive inclusive scan (ladder)
        for (int off = 1; off < 256; off <<= 1) {
            int add = (threadIdx.x >= (unsigned)off) ? buf[threadIdx.x - off] : 0;
            __syncthreads();
            buf[threadIdx.x] += add;
            __syncthreads();
        }
        int run = runningS;
        if (idx < N) {
            int excl = run + buf[threadIdx.x] - v;
            rowStart[idx] = excl;
            cursor[idx]   = excl;
        }
        __syncthreads();
        if (threadIdx.x == 255) runningS = run + buf[255];
        __syncthreads();
    }
    if (threadIdx.x == 0) rowStart[N] = runningS;   // == E
}

__global__ void fill_csr(const long long* __restrict__ src, const long long* __restrict__ dst,
                         int* __restrict__ cursor, int* __restrict__ edgeSrc, int E) {
    int e = blockIdx.x * blockDim.x + threadIdx.x;
    if (e >= E) return;
    int d = (int)dst[e];
    int pos = atomicAdd(&cursor[d], 1);
    edgeSrc[pos] = (int)src[e];
}

// ---------------- dense GEMM via f32 WMMA ----------------
// C[M x NC] = A[M x K] * B[K x NC], row-major, M and NC multiples of 16.
// One wave per 16x16 output tile; K zero-padded to a multiple of 4.
// Fragment layouts (ISA 7.12.2):
//   A 16x4 f32 : lane l<16 -> row=l,    v0=K0, v1=K1 ; lane l>=16 -> row=l-16, v0=K2, v1=K3
//   B 4x16 f32 : lane l    -> col=l%16, half=l/16,    v0=K(2*half), v1=K(2*half+1)
//   C/D 16x16  : vgpr i, lane l -> row = i + 8*(l>=16), col = l%16
template <int K, int NC>
__global__ void wmma_gemm_f32(const float* __restrict__ A,
                              const float* __restrict__ B,
                              float* __restrict__ C, int M) {
    const int wavesPerBlock = blockDim.x >> 5;
    const int wave = blockIdx.x * wavesPerBlock + ((int)threadIdx.x >> 5);
    constexpr int tilesN = NC >> 4;
    const int tilesM = M >> 4;
    if (wave >= tilesM * tilesN) return;          // wave-uniform exit: EXEC stays full

    const int tm   = wave / tilesN;
    const int tn   = wave % tilesN;
    const int lane = (int)threadIdx.x & 31;
    const int half = lane >> 4;
    const int l16  = lane & 15;

    const int arow = (tm << 4) + l16;
    const int bcol = (tn << 4) + l16;

    v8f c = {};
    #pragma unroll 8
    for (int k = 0; k < K; k += 4) {
        const int k0 = k + half * 2;
        const int k1 = k0 + 1;
        v2f a, b;
        a.x = (k0 < K) ? A[(size_t)arow * K + k0] : 0.0f;
        a.y = (k1 < K) ? A[(size_t)arow * K + k1] : 0.0f;
        b.x = (k0 < K) ? B[(size_t)k0 * NC + bcol] : 0.0f;
        b.y = (k1 < K) ? B[(size_t)k1 * NC + bcol] : 0.0f;
        c = __builtin_amdgcn_wmma_f32_16x16x4_f32(
                /*neg_a=*/false, a, /*neg_b=*/false, b,
                /*c_mod=*/(short)0, c, /*reuse_a=*/false, /*reuse_b=*/false);
    }

    const int rbase = (tm << 4) + half * 8;
    #pragma unroll
    for (int i = 0; i < 8; ++i)
        C[(size_t)(rbase + i) * NC + bcol] = c[i];
}

// ---------------- fused gather aggregation ----------------
// out[i][f] = relu( bias[f] + dis_i^2 * h[i][f] + sum_{e in in(i)} dis_i*dis_s * h[s][f] )
// One thread per (node, float4 chunk). For F=128 a node's 32 chunks are exactly
// one wave32, so the edge loop is wave-uniform (no divergence).
template <int F>
__global__ void gcn_aggregate(const float* __restrict__ h, const float* __restrict__ dis,
                              const int* __restrict__ rowStart, const int* __restrict__ edgeSrc,
                              const float* __restrict__ bias, float* __restrict__ out, int N) {
    constexpr int CHUNKS = F / 4;
    int t = blockIdx.x * blockDim.x + threadIdx.x;
    if (t >= N * CHUNKS) return;
    int i = t / CHUNKS;
    int c = t - i * CHUNKS;

    float di = dis[i];
    float4 hv = ((const float4*)(h + (size_t)i * F))[c];
    float wl = di * di;                                 // self-loop weight
    float4 acc;
    acc.x = wl * hv.x; acc.y = wl * hv.y; acc.z = wl * hv.z; acc.w = wl * hv.w;

    int e0 = rowStart[i], e1 = rowStart[i + 1];
    for (int e = e0; e < e1; ++e) {
        int s = edgeSrc[e];
        float w = di * dis[s];
        float4 v = ((const float4*)(h + (size_t)s * F))[c];
        acc.x += w * v.x; acc.y += w * v.y; acc.z += w * v.z; acc.w += w * v.w;
    }

    float4 bv = ((const float4*)bias)[c];
    acc.x += bv.x; acc.y += bv.y; acc.z += bv.z; acc.w += bv.w;
    acc.x = acc.x > 0.0f ? acc.x : 0.0f;
    acc.y = acc.y > 0.0f ? acc.y : 0.0f;
    acc.z = acc.z > 0.0f ? acc.z : 0.0f;
    acc.w = acc.w > 0.0f ? acc.w : 0.0f;
    ((float4*)(out + (size_t)i * F))[c] = acc;
}

// ---------------- pooling + FC head ----------------
__global__ void zero_pool(float* pool) {
    if (threadIdx.x < N_FEAT2) pool[threadIdx.x] = 0.0f;
}

__global__ void col_sum(const float* __restrict__ h, float* __restrict__ pool, int N) {
    int col = (int)threadIdx.x & (N_FEAT2 - 1);
    int rg  = (int)threadIdx.x >> 6;
    float s = 0.0f;
    for (int r = blockIdx.x * 4 + rg; r < N; r += gridDim.x * 4)
        s += h[(size_t)r * N_FEAT2 + col];
    atomicAdd(&pool[col], s);
}

__global__ void fc_head(const float* __restrict__ pool, const float* __restrict__ Wfc,
                        const float* __restrict__ bfc, float* __restrict__ out, float invN) {
    int j = (int)threadIdx.x;
    if (j < 3) {
        float acc = bfc[j];
        #pragma unroll
        for (int c = 0; c < N_FEAT2; ++c)
            acc += (pool[c] * invN) * Wfc[c * 3 + j];
        out[j] = acc;
    }
}

// ---------------------------------------------------------------------------
extern "C" void kernel_launch(void* const* d_in, const int* in_sizes, int n_in,
                              void* d_out, int out_size, void* d_ws, size_t ws_size,
                              hipStream_t stream) {
    (void)n_in; (void)out_size; (void)ws_size;

    const float*     x   = (const float*)d_in[0];
    const long long* ei  = (const long long*)d_in[1];
    const float*     W1  = (const float*)d_in[2];
    const float*     b1  = (const float*)d_in[3];
    const float*     W2  = (const float*)d_in[4];
    const float*     b2  = (const float*)d_in[5];
    const float*     Wfc = (const float*)d_in[6];
    const float*     bfc = (const float*)d_in[7];
    float*           out = (float*)d_out;

    const int N = in_sizes[0] / 3;      // 50000 (multiple of 16)
    const int E = in_sizes[1] / 2;      // 600000
    const long long* src = ei;
    const long long* dst = ei + E;

    // Workspace layout (all regions padded to 16B alignment):
    auto pad16 = [](size_t elems) { return (elems + 15) & ~(size_t)15; };
    char* p = (char*)d_ws;
    float* dis      = (float*)p; p += pad16(N) * 4;
    int*   cnt      = (int*)p;   p += pad16(N) * 4;
    int*   rowStart = (int*)p;   p += pad16(N + 1) * 4;
    int*   cursor   = (int*)p;   p += pad16(N) * 4;
    int*   edgeSrc  = (int*)p;   p += pad16(E) * 4;
    float* h1       = (float*)p; p += (size_t)N * N_FEAT1 * 4;   // reused below
    float* agg1     = (float*)p; p += (size_t)N * N_FEAT1 * 4;
    float* pool     = (float*)p;
    float* h2       = h1;                           // h1 dead after gemm2 reads agg1
    float* agg2     = h1 + (size_t)N * N_FEAT2;     // second half of h1 region

    const int B256 = 256;
    auto cdiv = [](int a, int b) { return (a + b - 1) / b; };

    // ---- CSR + normalization ----
    zero_cnt   <<<cdiv(N, B256), B256, 0, stream>>>(cnt, N);
    count_in   <<<cdiv(E, B256), B256, 0, stream>>>(dst, cnt, E);
    make_dis   <<<cdiv(N, B256), B256, 0, stream>>>(cnt, dis, N);
    scan_counts<<<1, B256, 0, stream>>>(cnt, rowStart, cursor, N);
    fill_csr   <<<cdiv(E, B256), B256, 0, stream>>>(src, dst, cursor, edgeSrc, E);

    // ---- layer 1 ----
    {
        const int tiles = (N >> 4) * (N_FEAT1 >> 4);          // 3125 * 8
        wmma_gemm_f32<3, N_FEAT1><<<cdiv(tiles, 8), B256, 0, stream>>>(x, W1, h1, N);
        gcn_aggregate<N_FEAT1><<<cdiv(N * (N_FEAT1 / 4), B256), B256, 0, stream>>>(
            h1, dis, rowStart, edgeSrc, b1, agg1, N);
    }

    // ---- layer 2 ----
    {
        const int tiles = (N >> 4) * (N_FEAT2 >> 4);          // 3125 * 4
        wmma_gemm_f32<N_FEAT1, N_FEAT2><<<cdiv(tiles, 8), B256, 0, stream>>>(agg1, W2, h2, N);
        gcn_aggregate<N_FEAT2><<<cdiv(N * (N_FEAT2 / 4), B256), B256, 0, stream>>>(
            h2, dis, rowStart, edgeSrc, b2, agg2, N);
    }

    // ---- mean pool + FC head ----
    zero_pool<<<1, 64, 0, stream>>>(pool);
    col_sum  <<<256, B256, 0, stream>>>(agg2, pool, N);
    fc_head  <<<1, 32, 0, stream>>>(pool, Wfc, bfc, out, 1.0f / (float)N);
}